// LocalCrossAttentionDown_41996190221024
// MI455X (gfx1250) — compile-verified
//
#include <hip/hip_runtime.h>
#include <hip/hip_bf16.h>
#include <math.h>

typedef __attribute__((ext_vector_type(2))) float v2f;
typedef __attribute__((ext_vector_type(8))) float v8f;

#define B_    4
#define H_    16
#define T_    4096
#define MAXN_ 64
#define R_    4
#define D_    64
#define P_    (MAXN_ * R_)
#define TILE  16

__global__ __launch_bounds__(32)
void lca_kernel(const float* __restrict__ qin,
                const float* __restrict__ xk,
                const float* __restrict__ xv,
                const int*   __restrict__ regions,
                const unsigned char* __restrict__ tmask,
                const unsigned char* __restrict__ nmask,
                float* __restrict__ out)
{
    __shared__ float sK[TILE][D_];   // 4 KB: K tile (RoPE'd in place)
    __shared__ float sV[TILE][D_];   // 4 KB: V tile
    __shared__ float sP[16][16];     // 1 KB: probs (A-layout staging)
    __shared__ float sQ[R_][D_];     // 1 KB: the 4 queries of this region

    const int lane  = threadIdx.x;         // 0..31 (wave32)
    const int gid   = blockIdx.x;          // B*H*MAXN = 4096
    const int nIdx  = gid & (MAXN_ - 1);   // region-1
    const int bh    = gid >> 6;            // b*H + h
    const int b     = bh / H_;
    const int n     = nIdx + 1;

    // ---- span of region n in sorted regions[b,:] (uniform binary searches) ----
    const int* reg = regions + b * T_;
    int lo = 0, hi = T_;
    while (lo < hi) { int mid = (lo + hi) >> 1; if (reg[mid] < n) lo = mid + 1; else hi = mid; }
    const int s = lo;
    lo = s; hi = T_;
    while (lo < hi) { int mid = (lo + hi) >> 1; if (reg[mid] < n + 1) lo = mid + 1; else hi = mid; }
    const int e = lo;

    const float* kbase = xk + (size_t)bh * T_ * D_;
    const float* vbase = xv + (size_t)bh * T_ * D_;
    const int    p0    = nIdx * R_;
    float*       obase = out + ((size_t)bh * P_ + p0) * D_;

    const bool pool_ok = (nmask[b * MAXN_ + nIdx] != 0);

    if (s >= e || !pool_ok) {
        // Fully-masked pool: softmax(-1e30 everywhere) is uniform over all T
        // -> output is the mean of all V rows. (Statistically never taken.)
        float2 acc = make_float2(0.f, 0.f);
        const float* vp = vbase + 2 * lane;
        for (int t = 0; t < T_; ++t) {
            float2 vv = *(const float2*)(vp + (size_t)t * D_);
            acc.x += vv.x; acc.y += vv.y;
        }
        const float invT = 1.0f / (float)T_;
        acc.x *= invT; acc.y *= invT;
        #pragma unroll
        for (int j = 0; j < R_; ++j) {
            obase[j * D_ + 2 * lane]     = acc.x;
            obase[j * D_ + 2 * lane + 1] = acc.y;
        }
        return;
    }

    // ---- stage Q (RoPE with pos=0 is identity) ----
    {
        const float* qb = qin + ((size_t)bh * P_ + p0) * D_;
        int r  = lane >> 3;           // 0..3
        int c0 = (lane & 7) * 8;      // 0..56
        float4 v0 = *(const float4*)(qb + r * D_ + c0);
        float4 v1 = *(const float4*)(qb + r * D_ + c0 + 4);
        *(float4*)&sQ[r][c0]     = v0;
        *(float4*)&sQ[r][c0 + 4] = v1;
    }
    __syncthreads();

    const int m16     = lane & 15;          // A-matrix row / P-row
    const int koff    = (lane >> 4) * 2;    // A/B K-pair select
    const int keyLane = lane & 15;          // B-matrix column
    const int part    = lane >> 4;

    // A-operand: Q in 16x4-chunk layout, rows 4..15 zero. 16 chunks cover D=64.
    v2f aQ[16];
    #pragma unroll
    for (int kk = 0; kk < 16; ++kk) {
        float2 t = *(const float2*)&sQ[m16 & 3][4 * kk + koff];
        bool on = (m16 < R_);
        aQ[kk].x = on ? t.x : 0.f;
        aQ[kk].y = on ? t.y : 0.f;
    }

    // inv_freq[i] = theta^(-i/32), split: this lane handles freqs part*16..part*16+15
    float invf[16];
    {
        const float c = -logf(10000.0f) / 32.0f;
        #pragma unroll
        for (int j = 0; j < 16; ++j)
            invf[j] = __expf((float)(part * 16 + j) * c);
    }

    float mRun[4], lRun[4];
    #pragma unroll
    for (int j = 0; j < 4; ++j) { mRun[j] = -1e30f; lRun[j] = 0.f; }

    v8f cO0 = {}, cO1 = {}, cO2 = {}, cO3 = {};
    const float scale = 0.125f;   // 1/sqrt(64)

    for (int t0 = s; t0 < e; t0 += TILE) {
        // ---- coalesced K/V tile -> LDS (each lane: half a row, 8 float4 per buf) ----
        {
            int r  = lane >> 1;
            int c0 = (lane & 1) * 32;
            int t  = t0 + r; if (t >= e) t = e - 1;   // clamp; masked later
            const float* kr = kbase + (size_t)t * D_ + c0;
            const float* vr = vbase + (size_t)t * D_ + c0;
            #pragma unroll
            for (int j = 0; j < 8; ++j) {
                float4 kd = *(const float4*)(kr + 4 * j);
                float4 vd = *(const float4*)(vr + 4 * j);
                *(float4*)&sK[r][c0 + 4 * j] = kd;
                *(float4*)&sV[r][c0 + 4 * j] = vd;
            }
            if (t0 + TILE < e) {
                __builtin_prefetch(kbase + (size_t)(t0 + TILE + r) * D_ + c0, 0, 1);
                __builtin_prefetch(vbase + (size_t)(t0 + TILE + r) * D_ + c0, 0, 1);
            }
        }
        __syncthreads();

        // ---- RoPE K in LDS: each (key, freq-half) owned by exactly one lane ----
        {
            int  key = keyLane;
            int  t   = t0 + key;
            float pos = (float)(((t < e) ? t : (e - 1)) - s + R_);
            #pragma unroll
            for (int j = 0; j < 16; ++j) {
                int   i = part * 16 + j;
                float ang = pos * invf[j];
                float sn, cs;
                __sincosf(ang, &sn, &cs);
                float k1 = sK[key][i];
                float k2 = sK[key][i + 32];
                sK[key][i]      = k1 * cs - k2 * sn;
                sK[key][i + 32] = k2 * cs + k1 * sn;
            }
        }
        __syncthreads();

        // ---- GEMM1: scores(16q x 16k) = Q · K^T via 16x f32 WMMA over D=64 ----
        v8f cS = {};
        #pragma unroll
        for (int kk = 0; kk < 16; ++kk) {
            float2 t = *(const float2*)&sK[keyLane][4 * kk + koff];
            v2f bK; bK.x = t.x; bK.y = t.y;
            cS = __builtin_amdgcn_wmma_f32_16x16x4_f32(false, aQ[kk], false, bK,
                                                       (short)0, cS, false, false);
        }

        // ---- online softmax over this tile (queries 0..3) ----
        int  tKey  = t0 + keyLane;
        int  tSafe = (tKey < e) ? tKey : (e - 1);
        bool valid = (lane < 16) && (tKey < e) && (tmask[b * T_ + tSafe] != 0);

        unsigned long long bal = __ballot(valid);
        if (bal == 0) { __syncthreads(); continue; }   // uniform

        float alpha[4];
        #pragma unroll
        for (int j = 0; j < 4; ++j) {
            float sc = valid ? cS[j] * scale : -1e30f;
            float mx = sc;                               // max over lanes 0..15
            mx = fmaxf(mx, __shfl_xor(mx, 1, 32));
            mx = fmaxf(mx, __shfl_xor(mx, 2, 32));
            mx = fmaxf(mx, __shfl_xor(mx, 4, 32));
            mx = fmaxf(mx, __shfl_xor(mx, 8, 32));
            mx = __shfl(mx, 0, 32);
            float mNew = fmaxf(mRun[j], mx);
            alpha[j] = __expf(mRun[j] - mNew);
            float p  = __expf(sc - mNew);                // ==0 for masked lanes
            float sum = p;
            sum += __shfl_xor(sum, 1, 32);
            sum += __shfl_xor(sum, 2, 32);
            sum += __shfl_xor(sum, 4, 32);
            sum += __shfl_xor(sum, 8, 32);
            sum = __shfl(sum, 0, 32);
            lRun[j] = lRun[j] * alpha[j] + sum;
            mRun[j] = mNew;
            sP[j + (part << 3)][keyLane] = p;            // rows 0..3 real, 8..11 zero
        }
        #pragma unroll
        for (int j = 4; j < 8; ++j)                       // rows 4..7 / 12..15 zero
            sP[j + (part << 3)][keyLane] = 0.f;
        __syncthreads();

        // ---- rescale output accumulators by alpha ----
        #pragma unroll
        for (int j = 0; j < 8; ++j) {
            float f = (j < 4) ? alpha[j] : 1.0f;          // zero rows: factor moot
            cO0[j] *= f; cO1[j] *= f; cO2[j] *= f; cO3[j] *= f;
        }

        // ---- GEMM2: O(16q x 64d) += P(16x16) · V(16x64), key-chunks of 4 ----
        #pragma unroll
        for (int kc = 0; kc < 4; ++kc) {
            float2 t = *(const float2*)&sP[m16][4 * kc + koff];
            v2f aP; aP.x = t.x; aP.y = t.y;
            int kr = 4 * kc + koff;
            v2f bV0, bV1, bV2, bV3;
            bV0.x = sV[kr][ 0 + keyLane]; bV0.y = sV[kr + 1][ 0 + keyLane];
            bV1.x = sV[kr][16 + keyLane]; bV1.y = sV[kr + 1][16 + keyLane];
            bV2.x = sV[kr][32 + keyLane]; bV2.y = sV[kr + 1][32 + keyLane];
            bV3.x = sV[kr][48 + keyLane]; bV3.y = sV[kr + 1][48 + keyLane];
            cO0 = __builtin_amdgcn_wmma_f32_16x16x4_f32(false, aP, false, bV0, (short)0, cO0, false, false);
            cO1 = __builtin_amdgcn_wmma_f32_16x16x4_f32(false, aP, false, bV1, (short)0, cO1, false, false);
            cO2 = __builtin_amdgcn_wmma_f32_16x16x4_f32(false, aP, false, bV2, (short)0, cO2, false, false);
            cO3 = __builtin_amdgcn_wmma_f32_16x16x4_f32(false, aP, false, bV3, (short)0, cO3, false, false);
        }
        __syncthreads();
    }

    // ---- epilogue: out[q][d] = O[q][d] / l[q]; C layout: VGPR j = query, lane = dim%16
    if (lane < 16) {
        #pragma unroll
        for (int j = 0; j < 4; ++j) {
            float inv = 1.0f / lRun[j];
            obase[(size_t)j * D_ +  0 + lane] = cO0[j] * inv;
            obase[(size_t)j * D_ + 16 + lane] = cO1[j] * inv;
            obase[(size_t)j * D_ + 32 + lane] = cO2[j] * inv;
            obase[(size_t)j * D_ + 48 + lane] = cO3[j] * inv;
        }
    }
}

extern "C" void kernel_launch(void* const* d_in, const int* in_sizes, int n_in,
                              void* d_out, int out_size, void* d_ws, size_t ws_size,
                              hipStream_t stream) {
    const float* q        = (const float*)d_in[0];
    const float* xk       = (const float*)d_in[1];
    const float* xv       = (const float*)d_in[2];
    const int*   regions  = (const int*)d_in[3];
    const unsigned char* tmask = (const unsigned char*)d_in[4];  // jnp bool = 1 byte
    const unsigned char* nmask = (const unsigned char*)d_in[5];  // jnp bool = 1 byte
    float* out = (float*)d_out;

    dim3 grid(B_ * H_ * MAXN_);   // 4096 waves: one per (b, h, region)
    hipLaunchKernelGGL(lca_kernel, grid, dim3(32), 0, stream,
                       q, xk, xv, regions, tmask, nmask, out);
}